// GNNEncoder_75788992905450
// MI455X (gfx1250) — compile-verified
//
#include <hip/hip_runtime.h>

// ---------------------------------------------------------------------------
// GNN encoder for MI455X (gfx1250, wave32).
//   h1 = relu( c_dst * SpMM( (feat*c_src) @ W1 ) + b1 )
//   h2 =       c_dst * SpMM( (h1 *c_src) @ W2 ) + b2
//   out = mean_n(h2)   -> [1, 64]
// GEMMs: V_WMMA_F32_16X16X4_F32 with both operands staged in LDS via the
// CDNA5 async copy path (GLOBAL_LOAD_ASYNC_TO_LDS_B128 + S_WAIT_ASYNCCNT).
// ---------------------------------------------------------------------------

#define NODES 100000
#define EDGES 1600000
#define KDIM  128      // IN == HID == 128
#define OUTC  64
#define ASTRIDE 132    // A-tile LDS row stride (floats): 16B-aligned, bank-spread

typedef __attribute__((ext_vector_type(2))) float v2f;
typedef __attribute__((ext_vector_type(8))) float v8f;

// ------------------------------------------------------- CDNA5 async helpers
// Copy 16 bytes from global address sbase plus goff to LDS offset ldsoff (GVS mode, per-lane)
__device__ __forceinline__ void async_copy_b128(unsigned ldsoff, unsigned goff,
                                                const void* sbase) {
    asm volatile("global_load_async_to_lds_b128 %0, %1, %2 offset:0"
                 :: "v"(ldsoff), "v"(goff), "s"(sbase)
                 : "memory");
}
__device__ __forceinline__ void wait_async0() {
    asm volatile("s_wait_asynccnt 0x0" ::: "memory");
}
__device__ __forceinline__ unsigned lds_byte_off(const void* p) {
    // addrspace(3) offset == low 32 bits of the flat shared address
    return (unsigned)(size_t)p;
}

// ---------------------------------------------------------------- utilities
__global__ void zero_f32(float* __restrict__ p, int n) {
    int i = blockIdx.x * blockDim.x + threadIdx.x;
    if (i < n) p[i] = 0.0f;
}

__global__ void degree_kernel(const int* __restrict__ src, const int* __restrict__ dst,
                              float* __restrict__ deg_src, float* __restrict__ deg_dst, int e) {
    int i = blockIdx.x * blockDim.x + threadIdx.x;
    if (i < e) {
        atomicAdd(&deg_src[src[i]], 1.0f);
        atomicAdd(&deg_dst[dst[i]], 1.0f);
    }
}

// in-place deg -> clip(deg,1)^-0.5  (c_src and c_dst jointly: 2N)
__global__ void coeff_kernel(float* __restrict__ d, int n) {
    int i = blockIdx.x * blockDim.x + threadIdx.x;
    if (i < n) {
        float v = d[i];
        v = v < 1.0f ? 1.0f : v;
        d[i] = 1.0f / sqrtf(v);
    }
}

// ---------------------------------------------------------------- WMMA GEMM
// out[rows, :] = diag(rowscale)*A @ W ; A:[M x 128], W:[128 x NCOL] row-major.
// Block: NCOL/16 waves, RTILES row-tiles of 16 rows each. W staged once in
// LDS per block; each A tile staged per row-tile. All staging is async.
template <int NCOL, int RTILES>
__global__ __launch_bounds__(NCOL * 2) void gemm_wmma(const float* __restrict__ A,
                                                      const float* __restrict__ W,
                                                      const float* __restrict__ rowscale,
                                                      float* __restrict__ out) {
    __shared__ float Wlds[KDIM * NCOL];
    __shared__ float Alds[16][ASTRIDE];

    constexpr int NT = NCOL * 2;
    const int t = threadIdx.x;

    // ---- stage whole W slab (128 x NCOL) asynchronously, once per block
    {
        const unsigned wbase = lds_byte_off(&Wlds[0]);
        constexpr int GROUPS = KDIM * NCOL / 4;        // float4 groups
#pragma unroll
        for (int g = t; g < GROUPS; g += NT)
            async_copy_b128(wbase + (unsigned)(g * 16), (unsigned)(g * 16), W);
    }

    const int lane = t & 31;
    const int wave = t >> 5;
    const int n0   = wave * 16;
    const int hsel = lane >> 4;   // 0: K pair {k0,k0+1}; 1: {k0+2,k0+3}
    const int mr   = lane & 15;   // A row owned by this lane
    const int nc   = lane & 15;   // B/D column within the 16-wide slab
    const unsigned abase = lds_byte_off(&Alds[0][0]);

    for (int tile = 0; tile < RTILES; ++tile) {
        const int row0 = (blockIdx.x * RTILES + tile) * 16;
        if (row0 >= NODES) break;

        __syncthreads();  // previous tile's Alds readers are done
        // ---- stage 16x128 A tile asynchronously (padded LDS rows)
        {
            constexpr int AG = 16 * KDIM / 4;          // 512 float4 groups
#pragma unroll
            for (int g = t; g < AG; g += NT) {
                const int r = g >> 5;                  // row in tile
                const int c4 = (g & 31) * 4;           // col (float index)
                async_copy_b128(abase + (unsigned)((r * ASTRIDE + c4) * 4),
                                (unsigned)(((row0 + r) * KDIM + c4) * 4), A);
            }
        }
        wait_async0();
        __syncthreads();  // W (first iter) + A visible to all waves

        const float rs = rowscale ? rowscale[row0 + mr] : 1.0f;

        v8f acc = {};
#pragma unroll 4
        for (int k0 = 0; k0 < KDIM; k0 += 4) {
            const int ka = k0 + 2 * hsel;
            v2f a, b;
            a.x = Alds[mr][ka] * rs;
            a.y = Alds[mr][ka + 1] * rs;
            b.x = Wlds[ka * NCOL + n0 + nc];
            b.y = Wlds[(ka + 1) * NCOL + n0 + nc];
            acc = __builtin_amdgcn_wmma_f32_16x16x4_f32(
                /*neg_a=*/false, a, /*neg_b=*/false, b,
                /*c_mod=*/(short)0, acc, /*reuse_a=*/false, /*reuse_b=*/false);
        }

        // D layout: VGPR r -> (M=r+8*hsel, N=nc) within the tile
        const int rbase = hsel * 8;
#pragma unroll
        for (int r = 0; r < 8; ++r)
            out[(row0 + rbase + r) * NCOL + n0 + nc] = acc[r];
    }
}

// ------------------------------------------------------- SpMM (edge scatter)
// m[dst[e], :] += h[src[e], :].  One thread per (edge, 4-feature chunk).
template <int F>
__global__ void aggregate_kernel(const int* __restrict__ src, const int* __restrict__ dst,
                                 const float* __restrict__ h, float* __restrict__ m, int e) {
    constexpr int CH = F / 4;
    const int t = blockIdx.x * blockDim.x + threadIdx.x;
    const int edge = t / CH;
    const int q = t - edge * CH;
    if (edge < e) {
        const int s = src[edge];
        const int d = dst[edge];
        const float4 v = *reinterpret_cast<const float4*>(h + s * F + q * 4);
        float* mp = m + d * F + q * 4;
        atomicAdd(mp + 0, v.x);
        atomicAdd(mp + 1, v.y);
        atomicAdd(mp + 2, v.z);
        atomicAdd(mp + 3, v.w);
    }
}

// layer-1 epilogue: m = relu(m*c_dst + b1) * c_src   (pre-scale for layer 2)
__global__ void post1_kernel(float* __restrict__ m, const float* __restrict__ c_dst,
                             const float* __restrict__ c_src, const float* __restrict__ b1,
                             int n) {
    int i = blockIdx.x * blockDim.x + threadIdx.x;
    if (i < n) {
        const int row = i >> 7;     // F = 128
        const int f = i & 127;
        float v = m[i] * c_dst[row] + b1[f];
        v = v > 0.0f ? v : 0.0f;
        m[i] = v * c_src[row];
    }
}

// column sums of m2*c_dst -> atomicAdd into out[64]
__global__ void reduce_kernel(const float* __restrict__ m2, const float* __restrict__ c_dst,
                              float* __restrict__ out, int rows_per_block, int mtot) {
    const int f = threadIdx.x;                   // blockDim = 64
    const int r0 = blockIdx.x * rows_per_block;
    int r1 = r0 + rows_per_block;
    if (r1 > mtot) r1 = mtot;
    float s = 0.0f;
    for (int r = r0; r < r1; ++r) s += m2[r * OUTC + f] * c_dst[r];
    atomicAdd(&out[f], s);
}

// out = out/N + b2
__global__ void finalize_kernel(float* __restrict__ out, const float* __restrict__ b2) {
    const int f = threadIdx.x;
    out[f] = out[f] * (1.0f / (float)NODES) + b2[f];
}

// ---------------------------------------------------------------- launcher
extern "C" void kernel_launch(void* const* d_in, const int* in_sizes, int n_in,
                              void* d_out, int out_size, void* d_ws, size_t ws_size,
                              hipStream_t stream) {
    const float* feat = (const float*)d_in[0];   // [N,128]
    const int*   src  = (const int*)d_in[1];     // [E]
    const int*   dst  = (const int*)d_in[2];     // [E]
    const float* W1   = (const float*)d_in[3];   // [128,128]
    const float* b1   = (const float*)d_in[4];   // [128]
    const float* W2   = (const float*)d_in[5];   // [128,64]
    const float* b2   = (const float*)d_in[6];   // [64]
    float* out = (float*)d_out;                  // [1,64]

    // workspace layout (floats): c_src | c_dst | bufA (N*128) | bufB (N*128)
    float* c_src = (float*)d_ws;
    float* c_dst = c_src + NODES;
    float* bufA  = c_dst + NODES;                // h1, later h2 (N*64 slice)
    float* bufB  = bufA + (size_t)NODES * KDIM;  // m1/x2, later m2

    const int TB = 256;
    const int RT = 4;                            // row-tiles per GEMM block
    const int nblk = (NODES / 16 + RT - 1) / RT;

    // 1) degrees -> normalization coefficients
    zero_f32<<<(2 * NODES + TB - 1) / TB, TB, 0, stream>>>(c_src, 2 * NODES);
    degree_kernel<<<(EDGES + TB - 1) / TB, TB, 0, stream>>>(src, dst, c_src, c_dst, EDGES);
    coeff_kernel<<<(2 * NODES + TB - 1) / TB, TB, 0, stream>>>(c_src, 2 * NODES);

    // 2) layer 1: h1 = (feat*c_src) @ W1   (WMMA fp32, async-staged LDS)
    gemm_wmma<128, 4><<<nblk, 256, 0, stream>>>(feat, W1, c_src, bufA);

    // 3) SpMM aggregate into m1, then fused epilogue
    zero_f32<<<(NODES * KDIM + TB - 1) / TB, TB, 0, stream>>>(bufB, NODES * KDIM);
    {
        const long long tot = (long long)EDGES * (KDIM / 4);
        aggregate_kernel<128><<<(int)((tot + TB - 1) / TB), TB, 0, stream>>>(src, dst, bufA, bufB, EDGES);
    }
    post1_kernel<<<(NODES * KDIM + TB - 1) / TB, TB, 0, stream>>>(bufB, c_dst, c_src, b1,
                                                                  NODES * KDIM);

    // 4) layer 2: h2 = x2 @ W2 (row scale already folded in)
    gemm_wmma<64, 4><<<nblk, 128, 0, stream>>>(bufB, W2, nullptr, bufA);

    // 5) SpMM aggregate into m2 (reuse bufB's first N*64)
    zero_f32<<<(NODES * OUTC + TB - 1) / TB, TB, 0, stream>>>(bufB, NODES * OUTC);
    {
        const long long tot = (long long)EDGES * (OUTC / 4);
        aggregate_kernel<64><<<(int)((tot + TB - 1) / TB), TB, 0, stream>>>(src, dst, bufA, bufB, EDGES);
    }

    // 6) mean over nodes: zero out, atomic column sums, finalize with /N + b2
    zero_f32<<<1, OUTC, 0, stream>>>(out, OUTC);
    {
        const int rows_per_block = 256;
        const int nb = (NODES + rows_per_block - 1) / rows_per_block;
        reduce_kernel<<<nb, OUTC, 0, stream>>>(bufB, c_dst, out, rows_per_block, NODES);
    }
    finalize_kernel<<<1, OUTC, 0, stream>>>(out, b2);
}